// EncoderForPretraining_47158740910835
// MI455X (gfx1250) — compile-verified
//
#include <hip/hip_runtime.h>
#include <hip/hip_bf16.h>
#include <stdint.h>

// Problem constants (match the reference).
constexpr int Bn = 64;       // batch
constexpr int Ln = 1500;     // embedding time
constexpr int Dn = 768;      // feature dim
constexpr int BIN = 160;     // raw steps per bin (T / L)
constexpr int NL = 10;       // L-chunks for deterministic two-stage reduction
constexpr int LCHUNK = Ln / NL;  // 150

typedef float v2f __attribute__((ext_vector_type(2)));
typedef float v4f __attribute__((ext_vector_type(4)));
typedef float v8f __attribute__((ext_vector_type(8)));

// ---------------------------------------------------------------------------
// Kernel 1: one wave per 160-byte bin of pad_mask. valid[b,l] = 1 iff no
// padded position in the bin. cnt[b] += 1 per valid bin (exact in f32).
// jax bool -> 1 byte per element; 160 bytes = 40 dwords, dword-aligned.
// pad_mask is streamed once -> non-temporal loads.
// ---------------------------------------------------------------------------
__global__ void mask_pool_kernel(const uint8_t* __restrict__ pad,
                                 float* __restrict__ valid,
                                 float* __restrict__ cnt) {
    int wid  = (blockIdx.x * blockDim.x + threadIdx.x) >> 5;  // bin id = b*L + l
    int lane = threadIdx.x & 31;
    if (wid >= Bn * Ln) return;  // wave-uniform
    const uint32_t* p = (const uint32_t*)(pad + (size_t)wid * BIN);
    uint32_t v = __builtin_nontemporal_load(&p[lane]);            // dwords 0..31
    if (lane < 8) v |= __builtin_nontemporal_load(&p[32 + lane]); // dwords 32..39
    int any = __any(v != 0u);
    if (lane == 0) {
        valid[wid] = any ? 0.0f : 1.0f;
        if (!any) atomicAdd(&cnt[wid / Ln], 1.0f);
    }
}

// ---------------------------------------------------------------------------
// Kernel 2: THE bandwidth kernel (~220-295 MB of embeddings). Block = 192
// threads; each lane owns 4 consecutive d (float4 -> global_load_b128, 16B
// per lane per instruction). valid[b,l] is block-uniform, so the skip branch
// is wave-uniform and genuinely avoids HBM traffic for padded bins (~25%).
// Embeddings are read exactly once and exceed L2 -> non-temporal loads.
// Partials are written once each (deterministic), summed in fixed order later.
// ---------------------------------------------------------------------------
__global__ void masked_sum_kernel(const float* __restrict__ emb,
                                  const float* __restrict__ valid,
                                  float* __restrict__ partial) {
    const int b   = blockIdx.y;
    const int lch = blockIdx.x;          // 0..NL-1
    const int t   = threadIdx.x;         // 0..191 -> d = 4t .. 4t+3
    const int l0  = lch * LCHUNK;
    const float* __restrict__ base = emb + ((size_t)(b * Ln + l0)) * Dn + 4 * t;
    const float* __restrict__ vrow = valid + b * Ln + l0;
    v4f s = {0.0f, 0.0f, 0.0f, 0.0f};
    for (int l = 0; l < LCHUNK; ++l) {
        float v = vrow[l];                     // uniform -> scalar load
        if (v > 0.0f) {                        // wave-uniform branch
            v4f x = __builtin_nontemporal_load((const v4f*)(base + (size_t)l * Dn));
            s += x;
        }
    }
    *(v4f*)(partial + ((size_t)(lch * Bn + b)) * Dn + 4 * t) = s;
}

// ---------------------------------------------------------------------------
// Kernel 3: mean[b,d] = (sum over NL partials, fixed order) / cnt[b]
// ---------------------------------------------------------------------------
__global__ void finalize_mean_kernel(const float* __restrict__ partial,
                                     const float* __restrict__ cnt,
                                     float* __restrict__ mean) {
    int idx = blockIdx.x * blockDim.x + threadIdx.x;  // b*Dn + d
    if (idx >= Bn * Dn) return;
    int b = idx / Dn;
    float s = 0.0f;
    #pragma unroll
    for (int c = 0; c < NL; ++c) s += partial[(size_t)c * Bn * Dn + idx];
    mean[idx] = s / cnt[b];
}

// ---------------------------------------------------------------------------
// WMMA f32 GEMM (NT): C[m,n] = sum_k A[m,k] * W[n,k], optional ReLU.
// One wave per 16x16 output tile, K-loop in steps of 4 using
// V_WMMA_F32_16X16X4_F32.
//   A frag (16x4 f32):  lane m = lane&15; float2 = A[m, k + ks], k+ks+1,
//                       ks = (lane>=16) ? 2 : 0            (ISA 7.12.2)
//   B frag (4x16):      VGPR i holds rows K=i and K=i+2 -> same ks pattern,
//                       column n = nbase + (lane&15)
//   C/D (v8f):          VGPR i -> row mbase + i + (lane>=16 ? 8 : 0),
//                       col nbase + (lane&15)
// ---------------------------------------------------------------------------
__global__ void gemm_nt_wmma(const float* __restrict__ A,
                             const float* __restrict__ W,
                             float* __restrict__ Cmat,
                             int Mtiles, int Ntiles, int K, int N, int doRelu) {
    const int tile = blockIdx.x * (blockDim.x >> 5) + (threadIdx.x >> 5);
    const int lane = threadIdx.x & 31;
    if (tile >= Mtiles * Ntiles) return;      // wave-uniform
    const int mt = tile % Mtiles;
    const int nt = tile / Mtiles;
    const int mbase = mt * 16, nbase = nt * 16;
    const int r  = lane & 15;
    const int ks = (lane >> 4) << 1;          // 0 or 2

    const float* __restrict__ arow = A + (size_t)(mbase + r) * K;
    const float* __restrict__ wrow = W + (size_t)(nbase + r) * K;

    v8f c = {};
    for (int k = 0; k < K; k += 4) {
        v2f a  = *(const v2f*)(arow + k + ks);   // 8B aligned (K even, ks even)
        v2f bb = *(const v2f*)(wrow + k + ks);
        c = __builtin_amdgcn_wmma_f32_16x16x4_f32(
                /*neg_a=*/false, a, /*neg_b=*/false, bb,
                /*c_mod=*/(short)0, c, /*reuse_a=*/false, /*reuse_b=*/false);
    }

    const int row0 = mbase + ((lane >> 4) << 3);
    const int col  = nbase + r;
    #pragma unroll
    for (int i = 0; i < 8; ++i) {
        float v = c[i];
        if (doRelu) v = fmaxf(v, 0.0f);
        Cmat[(size_t)(row0 + i) * N + col] = v;
    }
}

// ---------------------------------------------------------------------------
// Kernel 6: row-wise L2 normalize, one block per batch row.
// ---------------------------------------------------------------------------
__global__ void normalize_kernel(const float* __restrict__ outraw,
                                 float* __restrict__ out) {
    __shared__ float red[256];
    const int b = blockIdx.x, t = threadIdx.x;
    const float* row = outraw + (size_t)b * Dn;
    float s = 0.0f;
    for (int d = t; d < Dn; d += 256) { float v = row[d]; s += v * v; }
    red[t] = s;
    __syncthreads();
    for (int st = 128; st > 0; st >>= 1) {
        if (t < st) red[t] += red[t + st];
        __syncthreads();
    }
    const float inv = 1.0f / sqrtf(red[0]);
    for (int d = t; d < Dn; d += 256) out[(size_t)b * Dn + d] = row[d] * inv;
}

// ---------------------------------------------------------------------------
// Launch. Inputs: [0] embeddings f32 (B,L,D), [1] W1 f32 (2D,D),
// [2] W2 f32 (D,2D), [3] pad_mask bool (B,T). Output: f32 (B,D).
// Workspace (~3.2 MB): partial | cnt | valid | mean | h | outraw
// ---------------------------------------------------------------------------
extern "C" void kernel_launch(void* const* d_in, const int* in_sizes, int n_in,
                              void* d_out, int out_size, void* d_ws, size_t ws_size,
                              hipStream_t stream) {
    const float*   emb = (const float*)d_in[0];
    const float*   W1  = (const float*)d_in[1];
    const float*   W2  = (const float*)d_in[2];
    const uint8_t* pad = (const uint8_t*)d_in[3];
    float* out = (float*)d_out;

    float* partial = (float*)d_ws;                        // NL*B*D
    float* cnt     = partial + (size_t)NL * Bn * Dn;      // B
    float* valid   = cnt + Bn;                            // B*L
    float* mean    = valid + (size_t)Bn * Ln;             // B*D
    float* h       = mean + (size_t)Bn * Dn;              // B*2D
    float* outraw  = h + (size_t)Bn * 2 * Dn;             // B*D

    hipMemsetAsync(cnt, 0, Bn * sizeof(float), stream);

    // 1 wave per bin; 8 waves per 256-thread block.
    mask_pool_kernel<<<(Bn * Ln + 7) / 8, 256, 0, stream>>>(pad, valid, cnt);

    // NL l-chunks per batch row; 192 threads cover D=768 as float4 lanes.
    masked_sum_kernel<<<dim3(NL, Bn), 192, 0, stream>>>(emb, valid, partial);

    finalize_mean_kernel<<<(Bn * Dn + 255) / 256, 256, 0, stream>>>(partial, cnt, mean);

    // GEMM1: (64x768) @ W1^T -> h (64x1536), ReLU. 4x96 tiles, 4 waves/block.
    gemm_nt_wmma<<<(4 * 96) / 4, 128, 0, stream>>>(mean, W1, h, 4, 96, Dn, 2 * Dn, 1);

    // GEMM2: (64x1536) @ W2^T -> outraw (64x768). 4x48 tiles.
    gemm_nt_wmma<<<(4 * 48) / 4, 128, 0, stream>>>(h, W2, outraw, 4, 48, 2 * Dn, Dn, 0);

    normalize_kernel<<<Bn, 256, 0, stream>>>(outraw, out);
}